// Net_39694087750181
// MI455X (gfx1250) — compile-verified
//
#include <hip/hip_runtime.h>
#include <hip/hip_bf16.h>

typedef __attribute__((ext_vector_type(2))) float v2f;
typedef __attribute__((ext_vector_type(8))) float v8f;

#define BN_EPS 1e-5f

__device__ __forceinline__ v8f wmma_f32_16x16x4(v2f a, v2f b, v8f c) {
    return __builtin_amdgcn_wmma_f32_16x16x4_f32(false, a, false, b, (short)0, c,
                                                 false, false);
}

// ---------------------------------------------------------------------------
// Edge aggregation: AGG[dst] += X[src], F features per node (F = 128 or 32).
// One thread handles 4 consecutive features of one edge.
// ---------------------------------------------------------------------------
template <int F>
__global__ __launch_bounds__(256) void agg_kernel(const float* __restrict__ X,
                                                  const int* __restrict__ src,
                                                  const int* __restrict__ dst,
                                                  float* __restrict__ AGG,
                                                  int nedges) {
    constexpr int CH = F / 4;
    int t = blockIdx.x * blockDim.x + threadIdx.x;
    if (t >= nedges * CH) return;
    int e = t / CH;
    int ch = t - e * CH;
    int s = src[e];
    int d = dst[e];
    const float4 v = *(const float4*)(X + (long long)s * F + ch * 4);
    float* p = AGG + (long long)d * F + ch * 4;
    atomicAdd(p + 0, v.x);
    atomicAdd(p + 1, v.y);
    atomicAdd(p + 2, v.z);
    atomicAdd(p + 3, v.w);
}

// ---------------------------------------------------------------------------
// GIN MLP for one layer:
//   h  = X + AGG                          (16 rows per wave, K1 features)
//   ha = ReLU(h @ Wa[K1x32] + Ba)         (WMMA f32 16x16x4, two col tiles)
//   hb = ReLU(ha @ Wb[32x32] + Bb)        (WMMA via LDS-staged ha)
//   OUT = hb; colsum += per-col sum(hb); colsumsq += per-col sum(hb^2)
// One wave computes 16 rows; 8 waves per block. nrows divisible by 16.
// ---------------------------------------------------------------------------
template <int K1>
__global__ __launch_bounds__(256) void gin_mlp_kernel(
    const float* __restrict__ X, const float* __restrict__ AGG,
    const float* __restrict__ Wa, const float* __restrict__ Ba,
    const float* __restrict__ Wb, const float* __restrict__ Bb,
    float* __restrict__ OUT, float* __restrict__ colsum,
    float* __restrict__ colsumsq, int nrows) {
    __shared__ float lds_h[8][16 * 32];

    const int lane = threadIdx.x & 31;
    const int wid = threadIdx.x >> 5;
    const int wave = blockIdx.x * 8 + wid;
    const int row0 = wave * 16;
    if (row0 >= nrows) return;  // wave-uniform: EXEC all-ones for WMMA

    const int half = lane >> 4;  // 0: lanes 0-15 (K,K+1), 1: lanes 16-31 (K+2,K+3)
    const int l16 = lane & 15;
    const int arow = row0 + l16;  // A-matrix row owned by this lane

    // ---- GEMM1: (X+AGG)[16 x K1] @ Wa[K1 x 32] ----
    v8f c0 = {};
    v8f c1 = {};
    const float* xr = X + (long long)arow * K1;
    const float* gr = AGG + (long long)arow * K1;
    for (int k = 0; k < K1; k += 4) {
        const int ka = k + 2 * half;
        v2f a;
        a.x = xr[ka] + gr[ka];
        a.y = xr[ka + 1] + gr[ka + 1];
        v2f b0, b1;
        b0.x = Wa[ka * 32 + l16];
        b0.y = Wa[(ka + 1) * 32 + l16];
        b1.x = Wa[ka * 32 + 16 + l16];
        b1.y = Wa[(ka + 1) * 32 + 16 + l16];
        c0 = wmma_f32_16x16x4(a, b0, c0);
        c1 = wmma_f32_16x16x4(a, b1, c1);
    }

    // ---- bias + ReLU, stage ha into LDS (row-major 16x32 per wave) ----
    const float ba0 = Ba[l16];
    const float ba1 = Ba[16 + l16];
    float* myh = lds_h[wid];
#pragma unroll
    for (int r = 0; r < 8; ++r) {
        const int m = r + 8 * half;  // C/D layout: vgpr r, high half -> row r+8
        float h0 = c0[r] + ba0;
        float h1 = c1[r] + ba1;
        myh[m * 32 + l16] = h0 > 0.f ? h0 : 0.f;
        myh[m * 32 + 16 + l16] = h1 > 0.f ? h1 : 0.f;
    }
    // wave-local LDS visibility (stores before loads); no cross-wave sync needed
    asm volatile("s_wait_dscnt 0" ::: "memory");

    // ---- GEMM2: ha[16 x 32] @ Wb[32 x 32] ----
    v8f d0 = {};
    v8f d1 = {};
#pragma unroll
    for (int k = 0; k < 32; k += 4) {
        const int ka = k + 2 * half;
        v2f a;
        a.x = myh[l16 * 32 + ka];
        a.y = myh[l16 * 32 + ka + 1];
        v2f b0, b1;
        b0.x = Wb[ka * 32 + l16];
        b0.y = Wb[(ka + 1) * 32 + l16];
        b1.x = Wb[ka * 32 + 16 + l16];
        b1.y = Wb[(ka + 1) * 32 + 16 + l16];
        d0 = wmma_f32_16x16x4(a, b0, d0);
        d1 = wmma_f32_16x16x4(a, b1, d1);
    }

    // ---- bias + ReLU, store, accumulate BN statistics ----
    const float bb0 = Bb[l16];
    const float bb1 = Bb[16 + l16];
    float s0 = 0.f, s1 = 0.f, q0 = 0.f, q1 = 0.f;
#pragma unroll
    for (int r = 0; r < 8; ++r) {
        const int m = row0 + r + 8 * half;
        float h0 = d0[r] + bb0;
        float h1 = d1[r] + bb1;
        h0 = h0 > 0.f ? h0 : 0.f;
        h1 = h1 > 0.f ? h1 : 0.f;
        OUT[(long long)m * 32 + l16] = h0;
        OUT[(long long)m * 32 + 16 + l16] = h1;
        s0 += h0;
        q0 += h0 * h0;
        s1 += h1;
        q1 += h1 * h1;
    }
    atomicAdd(&colsum[l16], s0);
    atomicAdd(&colsum[16 + l16], s1);
    atomicAdd(&colsumsq[l16], q0);
    atomicAdd(&colsumsq[16 + l16], q1);
}

// ---------------------------------------------------------------------------
// BatchNorm with batch statistics: OUT = (H - mu) * rsqrt(var+eps) * g + be
// ---------------------------------------------------------------------------
__global__ __launch_bounds__(256) void bn_kernel(const float* __restrict__ H,
                                                 const float* __restrict__ colsum,
                                                 const float* __restrict__ colsumsq,
                                                 const float* __restrict__ g,
                                                 const float* __restrict__ be,
                                                 float* __restrict__ OUT, int n) {
    int i = blockIdx.x * blockDim.x + threadIdx.x;
    if (i >= n * 32) return;
    int col = i & 31;
    float invn = 1.0f / (float)n;
    float mu = colsum[col] * invn;
    float var = colsumsq[col] * invn - mu * mu;
    OUT[i] = (H[i] - mu) * rsqrtf(var + BN_EPS) * g[col] + be[col];
}

// ---------------------------------------------------------------------------
// Head: t = ReLU(X[16x32] @ Fw1 + Fb1) via WMMA, then per-row 32->2 projection.
// ---------------------------------------------------------------------------
__global__ __launch_bounds__(256) void head_kernel(
    const float* __restrict__ X, const float* __restrict__ Fw1,
    const float* __restrict__ Fb1, const float* __restrict__ Fw2,
    const float* __restrict__ Fb2, float* __restrict__ OUT, int nrows) {
    __shared__ float lds_t[8][16 * 32];

    const int lane = threadIdx.x & 31;
    const int wid = threadIdx.x >> 5;
    const int wave = blockIdx.x * 8 + wid;
    const int row0 = wave * 16;
    if (row0 >= nrows) return;

    const int half = lane >> 4;
    const int l16 = lane & 15;
    const int arow = row0 + l16;

    v8f c0 = {};
    v8f c1 = {};
    const float* xr = X + (long long)arow * 32;
#pragma unroll
    for (int k = 0; k < 32; k += 4) {
        const int ka = k + 2 * half;
        v2f a;
        a.x = xr[ka];
        a.y = xr[ka + 1];
        v2f b0, b1;
        b0.x = Fw1[ka * 32 + l16];
        b0.y = Fw1[(ka + 1) * 32 + l16];
        b1.x = Fw1[ka * 32 + 16 + l16];
        b1.y = Fw1[(ka + 1) * 32 + 16 + l16];
        c0 = wmma_f32_16x16x4(a, b0, c0);
        c1 = wmma_f32_16x16x4(a, b1, c1);
    }

    const float fb0 = Fb1[l16];
    const float fb1_ = Fb1[16 + l16];
    float* myt = lds_t[wid];
#pragma unroll
    for (int r = 0; r < 8; ++r) {
        const int m = r + 8 * half;
        float h0 = c0[r] + fb0;
        float h1 = c1[r] + fb1_;
        myt[m * 32 + l16] = h0 > 0.f ? h0 : 0.f;
        myt[m * 32 + 16 + l16] = h1 > 0.f ? h1 : 0.f;
    }
    asm volatile("s_wait_dscnt 0" ::: "memory");

    // final 32 -> 2 projection: lanes 0-15 each own one row
    if (lane < 16) {
        const int m = row0 + lane;
        float o0 = Fb2[0];
        float o1 = Fb2[1];
#pragma unroll
        for (int k = 0; k < 32; ++k) {
            float t = myt[lane * 32 + k];
            o0 += t * Fw2[k * 2 + 0];
            o1 += t * Fw2[k * 2 + 1];
        }
        OUT[(long long)m * 2 + 0] = o0;
        OUT[(long long)m * 2 + 1] = o1;
    }
}

// ---------------------------------------------------------------------------
// Host orchestration
// ---------------------------------------------------------------------------
extern "C" void kernel_launch(void* const* d_in, const int* in_sizes, int n_in,
                              void* d_out, int out_size, void* d_ws, size_t ws_size,
                              hipStream_t stream) {
    const float* x = (const float*)d_in[0];
    const int* ei = (const int*)d_in[1];  // (2, E): row 0 = src, row 1 = dst
    const float* w1a = (const float*)d_in[4];
    const float* b1a = (const float*)d_in[5];
    const float* w1b = (const float*)d_in[6];
    const float* b1b = (const float*)d_in[7];
    const float* g1 = (const float*)d_in[8];
    const float* be1 = (const float*)d_in[9];
    const float* w2a = (const float*)d_in[10];
    const float* b2a = (const float*)d_in[11];
    const float* w2b = (const float*)d_in[12];
    const float* b2b = (const float*)d_in[13];
    const float* g2 = (const float*)d_in[14];
    const float* be2 = (const float*)d_in[15];
    const float* w3a = (const float*)d_in[16];
    const float* b3a = (const float*)d_in[17];
    const float* w3b = (const float*)d_in[18];
    const float* b3b = (const float*)d_in[19];
    const float* g3 = (const float*)d_in[20];
    const float* be3 = (const float*)d_in[21];
    const float* fw1 = (const float*)d_in[22];
    const float* fb1 = (const float*)d_in[23];
    const float* fw2 = (const float*)d_in[24];
    const float* fb2 = (const float*)d_in[25];

    const int N = in_sizes[0] / 128;
    const int E = in_sizes[1] / 2;
    const int* src = ei;
    const int* dst = ei + E;

    // workspace layout (floats)
    float* agg = (float*)d_ws;                 // N*128 (reused as N*32 later)
    float* hbuf = agg + (size_t)N * 128;       // N*32
    float* xbuf = hbuf + (size_t)N * 32;       // N*32
    float* stats = xbuf + (size_t)N * 32;      // 64: [0..31]=sum, [32..63]=sumsq
    float* colsum = stats;
    float* colsumsq = stats + 32;

    float* out = (float*)d_out;

    const int waves = N / 16;  // N divisible by 16
    const int mlp_blocks = (waves + 7) / 8;
    const int bn_blocks = (N * 32 + 255) / 256;

    // ---- Layer 1 (F_IN = 128) ----
    hipMemsetAsync(agg, 0, (size_t)N * 128 * sizeof(float), stream);
    hipMemsetAsync(stats, 0, 64 * sizeof(float), stream);
    {
        int threads = E * 32;  // E * (128/4)
        agg_kernel<128><<<(threads + 255) / 256, 256, 0, stream>>>(x, src, dst,
                                                                   agg, E);
    }
    gin_mlp_kernel<128><<<mlp_blocks, 256, 0, stream>>>(
        x, agg, w1a, b1a, w1b, b1b, hbuf, colsum, colsumsq, N);
    bn_kernel<<<bn_blocks, 256, 0, stream>>>(hbuf, colsum, colsumsq, g1, be1,
                                             xbuf, N);

    // ---- Layer 2 (DIM = 32) ----
    hipMemsetAsync(agg, 0, (size_t)N * 32 * sizeof(float), stream);
    hipMemsetAsync(stats, 0, 64 * sizeof(float), stream);
    {
        int threads = E * 8;  // E * (32/4)
        agg_kernel<32><<<(threads + 255) / 256, 256, 0, stream>>>(xbuf, src, dst,
                                                                  agg, E);
    }
    gin_mlp_kernel<32><<<mlp_blocks, 256, 0, stream>>>(
        xbuf, agg, w2a, b2a, w2b, b2b, hbuf, colsum, colsumsq, N);
    bn_kernel<<<bn_blocks, 256, 0, stream>>>(hbuf, colsum, colsumsq, g2, be2,
                                             xbuf, N);

    // ---- Layer 3 (DIM = 32) ----
    hipMemsetAsync(agg, 0, (size_t)N * 32 * sizeof(float), stream);
    hipMemsetAsync(stats, 0, 64 * sizeof(float), stream);
    {
        int threads = E * 8;
        agg_kernel<32><<<(threads + 255) / 256, 256, 0, stream>>>(xbuf, src, dst,
                                                                  agg, E);
    }
    gin_mlp_kernel<32><<<mlp_blocks, 256, 0, stream>>>(
        xbuf, agg, w3a, b3a, w3b, b3b, hbuf, colsum, colsumsq, N);
    bn_kernel<<<bn_blocks, 256, 0, stream>>>(hbuf, colsum, colsumsq, g3, be3,
                                             xbuf, N);

    // ---- Head ----
    head_kernel<<<mlp_blocks, 256, 0, stream>>>(xbuf, fw1, fb1, fw2, fb2, out, N);
}